// OneHotEncoding_1778116460548
// MI455X (gfx1250) — compile-verified
//
#include <hip/hip_runtime.h>
#include <hip/hip_bf16.h>

// CDNA5 (gfx1250) wave32 WMMA nearest-neighbor.
// Score tile via V_WMMA_F32_16X16X4_F32:
//   A row (receiver r)   = (-2rx, -2ry, -2rz, 1)
//   B col (mesh point j) = ( mx,   my,   mz,  ||m||^2 )
//   D[r][j] = ||m||^2 - 2 r.m   (same argmin as true squared distance)

typedef __attribute__((ext_vector_type(2))) float v2f;
typedef __attribute__((ext_vector_type(8))) float v8f;

#define RB 4  // 64 receivers / 16 rows per WMMA

__device__ __forceinline__ unsigned fkey(float v) {
    // monotone float -> uint mapping (min over uint == min over float)
    unsigned u = __float_as_uint(v);
    return (u & 0x80000000u) ? ~u : (u | 0x80000000u);
}

__global__ void nn_init(unsigned long long* __restrict__ ws, int B) {
    int i = blockIdx.x * blockDim.x + threadIdx.x;
    if (i < B) ws[i] = 0xFFFFFFFFFFFFFFFFull;
}

// One 16-mesh-point tile: GUARD=false -> branch-free fast path (tile fully
// in-range), GUARD=true -> per-lane bounds handling (last partial tile only).
template <bool GUARD>
__device__ __forceinline__ void process_tile(
        int t, int L, int mrow, int half,
        const float* __restrict__ mesh, float* __restrict__ out,
        const v2f (&a)[RB], float (&minv)[RB][8], unsigned (&mini)[RB][8]) {
    const int  j     = (t << 4) + mrow;
    const bool valid = !GUARD || (j < L);

    float mx = 0.0f, my = 0.0f, mz = 0.0f, msq = 3.4e38f;
    if (valid) {
        mx  = mesh[3 * (size_t)j + 0];
        my  = mesh[3 * (size_t)j + 1];
        mz  = mesh[3 * (size_t)j + 2];
        msq = __fmaf_rn(mx, mx, __fmaf_rn(my, my, mz * mz));
    }
    if (valid && half == 0) {
        // fused (L,4) output row; one-hot column placeholder 0 (b128 store)
        float4 row = make_float4(mx, my, mz, 0.0f);
        *(float4*)(out + 4 * (size_t)j) = row;
    }

    // B matrix 4x16: VGPR0 = K0/K2 lane halves, VGPR1 = K1/K3 lane halves
    v2f b;
    b.x = half ? mz  : mx;
    b.y = half ? msq : my;

    const v8f czero = {};
#pragma unroll
    for (int rb = 0; rb < RB; ++rb) {
        v8f d = __builtin_amdgcn_wmma_f32_16x16x4_f32(
                    false, a[rb], false, b, (short)0, czero, false, false);
#pragma unroll
        for (int v = 0; v < 8; ++v) {
            float s = d[v];
            bool  lt = GUARD ? (valid && (s < minv[rb][v])) : (s < minv[rb][v]);
            if (lt) {               // -> v_cmp + v_dual_cndmask, no branch
                minv[rb][v] = s;
                mini[rb][v] = (unsigned)j;
            }
        }
    }
}

__global__ __launch_bounds__(256) void nn_main(
        const float* __restrict__ mesh,      // (L,3)
        const float* __restrict__ recv,      // (B,3), B assumed 64
        float* __restrict__ out,             // (L,4) section of d_out
        unsigned long long* __restrict__ ws, // 64 packed (key,idx) slots
        int L, int B) {
    const int lane   = threadIdx.x & 31;
    const int mrow   = lane & 15;   // M (for A) / N (for B,D) position
    const int half   = lane >> 4;   // lane half selects K pair / M+8
    const int wave   = (int)((blockIdx.x * blockDim.x + threadIdx.x) >> 5);
    const int nwaves = (int)((gridDim.x * blockDim.x) >> 5);

    // --- A matrices (per 16-receiver row block), loaded once ------------
    // 32-bit A 16x4 layout: VGPR0 = K0 (lanes 0-15) / K2 (lanes 16-31),
    //                       VGPR1 = K1 (lanes 0-15) / K3 (lanes 16-31).
    v2f a[RB];
#pragma unroll
    for (int rb = 0; rb < RB; ++rb) {
        int r = rb * 16 + mrow;
        if (r >= B) r = B - 1;      // safety clamp (B is 64 in practice)
        float rx = recv[3 * r + 0];
        float ry = recv[3 * r + 1];
        float rz = recv[3 * r + 2];
        a[rb].x = half ? (-2.0f * rz) : (-2.0f * rx);
        a[rb].y = half ? 1.0f        : (-2.0f * ry);
    }

    // --- per-lane running argmin state: 4 row blocks x 8 D VGPRs ---------
    float    minv[RB][8];
    unsigned mini[RB][8];
#pragma unroll
    for (int rb = 0; rb < RB; ++rb)
#pragma unroll
        for (int v = 0; v < 8; ++v) {
            minv[rb][v] = __int_as_float(0x7f800000); // +inf
            mini[rb][v] = 0u;
        }

    const int full_tiles = L >> 4;            // tiles with all 16 points valid
    const bool has_tail  = (L & 15) != 0;

    // Branch-free hot loop over full tiles (wave-uniform trip count).
    for (int t = wave; t < full_tiles; t += nwaves)
        process_tile<false>(t, L, mrow, half, mesh, out, a, minv, mini);

    // At most one partial tile; give it to exactly one wave.
    if (has_tail && (full_tiles % nwaves) == wave)
        process_tile<true>(full_tiles, L, mrow, half, mesh, out, a, minv, mini);

    // --- cross-lane reduction: 16 lanes per half share a receiver --------
#pragma unroll
    for (int rb = 0; rb < RB; ++rb) {
#pragma unroll
        for (int v = 0; v < 8; ++v) {
            float    mv = minv[rb][v];
            unsigned mi = mini[rb][v];
#pragma unroll
            for (int off = 8; off >= 1; off >>= 1) {
                float    ov = __shfl_xor(mv, off, 32);
                unsigned oi = (unsigned)__shfl_xor((int)mi, off, 32);
                if (ov < mv || (ov == mv && oi < mi)) { mv = ov; mi = oi; }
            }
            if (mrow == 0) {
                int r = rb * 16 + half * 8 + v;   // D layout: M = v + 8*half
                if (r < B) {
                    unsigned long long key =
                        ((unsigned long long)fkey(mv) << 32) | (unsigned long long)mi;
                    atomicMin(&ws[r], key);       // global_atomic_min_u64
                }
            }
        }
    }
}

__global__ void nn_final(const float* __restrict__ mesh,
                         const unsigned long long* __restrict__ ws,
                         float* __restrict__ out, int L, int B) {
    int r = blockIdx.x * blockDim.x + threadIdx.x;
    if (r < B) {
        unsigned idx = (unsigned)(ws[r] & 0xFFFFFFFFull);
        if (idx >= (unsigned)L) idx = 0;
        float x = mesh[3 * (size_t)idx + 0];
        float y = mesh[3 * (size_t)idx + 1];
        float z = mesh[3 * (size_t)idx + 2];
        float* cp = out + 4 * (size_t)L + 3 * (size_t)r;
        cp[0] = x; cp[1] = y; cp[2] = z;
        out[4 * (size_t)idx + 3] = 1.0f;   // scatter one-hot (idempotent on ties)
    }
}

extern "C" void kernel_launch(void* const* d_in, const int* in_sizes, int n_in,
                              void* d_out, int out_size, void* d_ws, size_t ws_size,
                              hipStream_t stream) {
    const float* mesh = (const float*)d_in[0];
    const float* recv = (const float*)d_in[1];
    const int L = in_sizes[0] / 3;
    const int B = in_sizes[1] / 3;
    float* out = (float*)d_out;
    unsigned long long* ws = (unsigned long long*)d_ws;

    nn_init<<<(B + 63) / 64, 64, 0, stream>>>(ws, B);
    // 1024 blocks x 8 wave32 = 8192 waves; ~31 sixteen-point tiles each.
    nn_main<<<1024, 256, 0, stream>>>(mesh, recv, out, ws, L, B);
    nn_final<<<(B + 63) / 64, 64, 0, stream>>>(mesh, ws, out, L, B);
}